// CrossAttentionFusion_14087492731535
// MI455X (gfx1250) — compile-verified
//
#include <hip/hip_runtime.h>
#include <hip/hip_bf16.h>

// ---------------------------------------------------------------------------
// CrossAttentionFusion, KV-len == 1  =>  softmax over axis of size 1 == 1.0
// exactly, so:  out[b,n,:] = v[b,:] @ Wp + bp  (q / Wq / image_patches dead)
// with          v = (cnn @ Wkv)[:, C:2C]
//
// Shapes: B=64, N=576, C=768, CNN=2048.
// Work:   two tiny f32 GEMMs (V_WMMA_F32_16X16X4_F32) + 113MB broadcast store
//         (store-bandwidth bound: ~4.9us at 23.3 TB/s HBM).
// ---------------------------------------------------------------------------

typedef __attribute__((ext_vector_type(2))) float v2f;
typedef __attribute__((ext_vector_type(4))) float v4f;
typedef __attribute__((ext_vector_type(8))) float v8f;

// Generic small GEMM: C[M,N] = A[M,K] * B[K,N] (+ bias[N]), all f32 row-major.
// One wave32 computes one 16x16 tile with V_WMMA_F32_16X16X4_F32.
//   A frag (16x4, ISA 7.12.2): lanes 0-15 -> M=lane, VGPR0=K0,VGPR1=K1;
//                              lanes 16-31 -> M=lane-16, VGPR0=K2,VGPR1=K3.
//   B frag (4x16) assumed symmetric: VGPR0 = rows {K0,K2}, VGPR1 = rows {K1,K3}.
//   C/D (16x16): VGPR r -> row r (lanes 0-15) and row r+8 (lanes 16-31).
__global__ __launch_bounds__(128) void wmma_gemm_f32_kernel(
    const float* __restrict__ A, int lda,
    const float* __restrict__ B, int ldb,
    const float* __restrict__ bias,
    float* __restrict__ C, int ldc,
    int tilesN, int K)
{
    const int lane = threadIdx.x & 31;
    const int wave = threadIdx.x >> 5;
    const int tile = blockIdx.x * 4 + wave;       // 4 waves per block
    const int tm   = tile / tilesN;
    const int tn   = tile % tilesN;

    const int m   = (lane & 15);                  // A row within tile
    const int n   = (lane & 15);                  // B/C col within tile
    const int kh  = (lane >> 4) << 1;             // K sub-offset: 0 or 2

    const float* Arow = A + (size_t)(tm * 16 + m) * lda + kh;
    const float* Bcol = B + (size_t)kh * ldb + (tn * 16 + n);

    v8f acc = {};
    #pragma unroll 4
    for (int k = 0; k < K; k += 4) {
        v2f a = *(const v2f*)(Arow + k);          // A[m][k+kh], A[m][k+kh+1]
        v2f b;
        b.x = Bcol[(size_t)k * ldb];              // B[k+kh  ][n]
        b.y = Bcol[(size_t)(k + 1) * ldb];        // B[k+kh+1][n]
        acc = __builtin_amdgcn_wmma_f32_16x16x4_f32(
                  /*neg_a=*/false, a, /*neg_b=*/false, b,
                  /*c_mod=*/(short)0, acc, /*reuse_a=*/false, /*reuse_b=*/false);
    }

    const int   col   = tn * 16 + n;
    const int   rbase = tm * 16 + ((lane >> 4) << 3);   // rows rbase..rbase+7
    const float bv    = bias ? bias[col] : 0.0f;
    #pragma unroll
    for (int r = 0; r < 8; ++r) {
        C[(size_t)(rbase + r) * ldc + col] = acc[r] + bv;
    }
}

// Broadcast y[64,768] -> out[64,576,768]. float4 non-temporal stores,
// each thread writes 8 consecutive n-rows for one (b, c4).
__global__ __launch_bounds__(256) void bcast_rows_kernel(
    const v4f* __restrict__ y, v4f* __restrict__ out)
{
    constexpr int C4 = 768 / 4;   // 192
    constexpr int N  = 576;
    constexpr int NC = N / 8;     // 72

    int idx = blockIdx.x * blockDim.x + threadIdx.x;   // [0, 64*72*192)
    int c4  = idx % C4;
    int t   = idx / C4;
    int nch = t % NC;
    int b   = t / NC;

    v4f val = y[(size_t)b * C4 + c4];
    v4f* o  = out + ((size_t)b * N + (size_t)nch * 8) * C4 + c4;
    #pragma unroll
    for (int i = 0; i < 8; ++i) {
        __builtin_nontemporal_store(val, o);
        o += C4;
    }
}

extern "C" void kernel_launch(void* const* d_in, const int* in_sizes, int n_in,
                              void* d_out, int out_size, void* d_ws, size_t ws_size,
                              hipStream_t stream) {
    (void)in_sizes; (void)n_in; (void)out_size; (void)ws_size;

    // setup_inputs order: image_patches, cnn_feature_vector, Wq, Wkv, Wp, bp
    const float* cnn = (const float*)d_in[1];   // [64, 2048]
    const float* Wkv = (const float*)d_in[3];   // [2048, 1536]
    const float* Wp  = (const float*)d_in[4];   // [768, 768]
    const float* bp  = (const float*)d_in[5];   // [768]
    float*       out = (float*)d_out;           // [64, 576, 768]

    float* v = (float*)d_ws;                    // [64, 768]
    float* y = v + 64 * 768;                    // [64, 768]

    // v = cnn @ Wkv[:, 768:1536]   (M=64,N=768,K=2048) -> 4x48 tiles = 192 waves
    wmma_gemm_f32_kernel<<<48, 128, 0, stream>>>(
        cnn, 2048, Wkv + 768, 1536, nullptr, v, 768, /*tilesN=*/48, /*K=*/2048);

    // y = v @ Wp + bp              (M=64,N=768,K=768)
    wmma_gemm_f32_kernel<<<48, 128, 0, stream>>>(
        v, 768, Wp, 768, bp, y, 768, /*tilesN=*/48, /*K=*/768);

    // out[b,n,:] = y[b,:]  — 113MB of coalesced NT float4 stores
    const int threads = 64 * (576 / 8) * (768 / 4);    // 884736
    bcast_rows_kernel<<<threads / 256, 256, 0, stream>>>(
        (const v4f*)y, (v4f*)out);
}